// GATLayer_19172734010016
// MI455X (gfx1250) — compile-verified
//
#include <hip/hip_runtime.h>
#include <hip/hip_bf16.h>
#include <math.h>

#define N_NODES 100000
#define IN_CH   128
#define OUT_CH  32
#define HEADS   4
#define HC      128          // HEADS*OUT_CH
#define NEG_SLOPE 0.2f

typedef __attribute__((ext_vector_type(2))) float v2f;
typedef __attribute__((ext_vector_type(8))) float v8f;

// ---- monotone float <-> uint mapping for atomicMax on floats ----
__device__ __forceinline__ unsigned floatFlip(float f) {
    unsigned u = __float_as_uint(f);
    return (u & 0x80000000u) ? ~u : (u | 0x80000000u);
}
__device__ __forceinline__ float floatUnflip(unsigned u) {
    return __uint_as_float((u & 0x80000000u) ? (u & 0x7fffffffu) : ~u);
}
#define FLIP_NEG_INF 0x007FFFFFu   // floatFlip(-inf)

// ---------------- init: zero agg + sums, seed per-(node,head) max ----------------
__global__ void init_kernel(float* __restrict__ agg, float* __restrict__ sumbuf,
                            unsigned* __restrict__ mbuf) {
    int i = blockIdx.x * blockDim.x + threadIdx.x;
    if (i < N_NODES * HC) agg[i] = 0.0f;
    if (i < N_NODES * HEADS) { sumbuf[i] = 0.0f; mbuf[i] = FLIP_NEG_INF; }
}

// ---------------- GEMM1: h[N,128] = x[N,128] @ W[128,128]^T via WMMA f32 ----------------
// one wave per 16x16 output tile; K-loop of v_wmma_f32_16x16x4_f32
__global__ void __launch_bounds__(256) gemm_x_proj(const float* __restrict__ x,
                                                   const float* __restrict__ W,
                                                   float* __restrict__ h) {
    const int wave = threadIdx.x >> 5;
    const int lane = threadIdx.x & 31;
    const int tile = blockIdx.x * 8 + wave;        // 6250*8 = 50000 tiles exactly
    const int nT = tile & 7;                       // 8 tiles of 16 output channels
    const int mT = tile >> 3;                      // 6250 tiles of 16 rows
    const int laneLo = lane & 15;
    const int kOff = (lane < 16) ? 0 : 2;          // A/B frag: lanes 0-15 hold K0,K1; 16-31 hold K2,K3
    const size_t row = (size_t)(mT * 16 + laneLo);
    const size_t col = (size_t)(nT * 16 + laneLo);

    const float* aPtr = x + row * IN_CH + kOff;    // A[m][k], row-major
    const float* bPtr = W + col * IN_CH + kOff;    // B[k][n] = W[n][k], W row-major

    v8f acc = {};
    #pragma unroll 8
    for (int k = 0; k < IN_CH; k += 4) {
        v2f a = *(const v2f*)(aPtr + k);
        v2f b = *(const v2f*)(bPtr + k);
        acc = __builtin_amdgcn_wmma_f32_16x16x4_f32(false, a, false, b,
                                                    (short)0, acc, false, false);
    }

    // D layout: lanes0-15 N=lane,M=r ; lanes16-31 N=lane-16,M=r+8
    const size_t drow0 = (size_t)(mT * 16 + ((lane < 16) ? 0 : 8));
    float* out = h + drow0 * HC + nT * 16 + laneLo;
    #pragma unroll
    for (int r = 0; r < 8; ++r) out[(size_t)r * HC] = acc[r];
}

// ---------------- per-(node,head) attention logits ----------------
__global__ void node_attn(const float* __restrict__ h,
                          const float* __restrict__ att_src,
                          const float* __restrict__ att_dst,
                          float* __restrict__ a_src_n, float* __restrict__ a_dst_n) {
    int t = blockIdx.x * blockDim.x + threadIdx.x;
    if (t >= N_NODES * HEADS) return;
    int n = t >> 2, head = t & 3;
    const float* hp = h + (size_t)n * HC + head * OUT_CH;
    const float* as = att_src + head * OUT_CH;
    const float* ad = att_dst + head * OUT_CH;
    float ss = 0.f, sd = 0.f;
    #pragma unroll
    for (int c = 0; c < OUT_CH; ++c) { float v = hp[c]; ss += v * as[c]; sd += v * ad[c]; }
    a_src_n[t] = ss;
    a_dst_n[t] = sd;
}

// ---------------- edge pass 1: leaky-relu logits + segment max ----------------
__global__ void edge_logits_max(const int* __restrict__ ei, int E, int Etot,
                                const float* __restrict__ a_src_n,
                                const float* __restrict__ a_dst_n,
                                float* __restrict__ ebuf, unsigned* __restrict__ mbuf) {
    int idx = blockIdx.x * blockDim.x + threadIdx.x;
    if (idx >= Etot) return;
    int s, d;
    if (idx < E) { s = ei[idx]; d = ei[E + idx]; } else { s = d = idx - E; }
    const float* asp = a_src_n + (size_t)s * HEADS;
    const float* adp = a_dst_n + (size_t)d * HEADS;
    #pragma unroll
    for (int hh = 0; hh < HEADS; ++hh) {
        float e = asp[hh] + adp[hh];
        e = (e > 0.f) ? e : NEG_SLOPE * e;
        ebuf[(size_t)idx * HEADS + hh] = e;
        atomicMax(mbuf + (size_t)d * HEADS + hh, floatFlip(e));
    }
}

// ---------------- edge pass 2: exp(e - max) + segment sum ----------------
__global__ void edge_exp_sum(const int* __restrict__ ei, int E, int Etot,
                             const unsigned* __restrict__ mbuf,
                             float* __restrict__ ebuf, float* __restrict__ sumbuf) {
    int idx = blockIdx.x * blockDim.x + threadIdx.x;
    if (idx >= Etot) return;
    int d = (idx < E) ? ei[E + idx] : (idx - E);
    #pragma unroll
    for (int hh = 0; hh < HEADS; ++hh) {
        float m = floatUnflip(mbuf[(size_t)d * HEADS + hh]);
        float w = __expf(ebuf[(size_t)idx * HEADS + hh] - m);
        ebuf[(size_t)idx * HEADS + hh] = w;
        atomicAdd(sumbuf + (size_t)d * HEADS + hh, w);
    }
}

// ---------------- edge pass 3: alpha = w / sum[dst] (in place) ----------------
__global__ void edge_alpha(const int* __restrict__ ei, int E, int Etot,
                           const float* __restrict__ sumbuf, float* __restrict__ ebuf) {
    int idx = blockIdx.x * blockDim.x + threadIdx.x;
    if (idx >= Etot) return;
    int d = (idx < E) ? ei[E + idx] : (idx - E);
    #pragma unroll
    for (int hh = 0; hh < HEADS; ++hh)
        ebuf[(size_t)idx * HEADS + hh] /= sumbuf[(size_t)d * HEADS + hh];
}

// ---------------- edge pass 4: scatter alpha * h[src] (float4 per thread) ----------------
__global__ void edge_aggregate(const int* __restrict__ ei, int E, int Etot,
                               const float* __restrict__ ebuf,
                               const float* __restrict__ h, float* __restrict__ agg) {
    long long t = (long long)blockIdx.x * blockDim.x + threadIdx.x;
    const long long total = (long long)Etot * (HC / 4);
    if (t >= total) return;
    int idx = (int)(t >> 5);           // 32 float4-groups per edge
    int ch  = ((int)(t & 31)) << 2;    // channel base
    int head = ch >> 5;                // OUT_CH == 32
    int s, d;
    if (idx < E) { s = ei[idx]; d = ei[E + idx]; } else { s = d = idx - E; }
    float alpha = ebuf[(size_t)idx * HEADS + head];
    const float4 hv = *(const float4*)(h + (size_t)s * HC + ch);
    float* out = agg + (size_t)d * HC + ch;
    atomicAdd(out + 0, alpha * hv.x);
    atomicAdd(out + 1, alpha * hv.y);
    atomicAdd(out + 2, alpha * hv.z);
    atomicAdd(out + 3, alpha * hv.w);
}

// ---------------- GEMM2 + bias + ELU: out[N,32] = elu((agg+bias) @ W_lin^T + b_lin) ----------------
__global__ void __launch_bounds__(256) gemm_out_elu(const float* __restrict__ agg,
                                                    const float* __restrict__ bias_conv,
                                                    const float* __restrict__ W_lin,
                                                    const float* __restrict__ b_lin,
                                                    float* __restrict__ out) {
    const int wave = threadIdx.x >> 5;
    const int lane = threadIdx.x & 31;
    const int tile = blockIdx.x * 8 + wave;        // 6250*2 = 12500 tiles
    if (tile >= (N_NODES / 16) * 2) return;        // wave-uniform: EXEC stays all-ones
    const int nT = tile & 1;
    const int mT = tile >> 1;
    const int laneLo = lane & 15;
    const int kOff = (lane < 16) ? 0 : 2;
    const size_t row = (size_t)(mT * 16 + laneLo);
    const size_t col = (size_t)(nT * 16 + laneLo);

    const float* aPtr = agg + row * HC;
    const float* bPtr = W_lin + col * HC + kOff;

    v8f acc = {};
    #pragma unroll 8
    for (int k = 0; k < HC; k += 4) {
        int ka = k + kOff;
        v2f a; a.x = aPtr[ka] + bias_conv[ka];
               a.y = aPtr[ka + 1] + bias_conv[ka + 1];
        v2f b = *(const v2f*)(bPtr + k);
        acc = __builtin_amdgcn_wmma_f32_16x16x4_f32(false, a, false, b,
                                                    (short)0, acc, false, false);
    }

    const size_t drow0 = (size_t)(mT * 16 + ((lane < 16) ? 0 : 8));
    const int dcol = nT * 16 + laneLo;
    const float bl = b_lin[dcol];
    #pragma unroll
    for (int r = 0; r < 8; ++r) {
        float v = acc[r] + bl;
        v = (v > 0.f) ? v : expm1f(v);             // ELU, alpha = 1
        out[(drow0 + (size_t)r) * OUT_CH + dcol] = v;
    }
}

extern "C" void kernel_launch(void* const* d_in, const int* in_sizes, int n_in,
                              void* d_out, int out_size, void* d_ws, size_t ws_size,
                              hipStream_t stream) {
    const float* x        = (const float*)d_in[0];
    const int*   ei       = (const int*)d_in[1];
    const float* W        = (const float*)d_in[2];
    const float* att_src  = (const float*)d_in[3];
    const float* att_dst  = (const float*)d_in[4];
    const float* bias_conv= (const float*)d_in[5];
    const float* W_lin    = (const float*)d_in[6];
    const float* b_lin    = (const float*)d_in[7];
    float* out = (float*)d_out;

    const int E    = in_sizes[1] / 2;
    const int Etot = E + N_NODES;

    // workspace layout (floats): h | a_src | a_dst | max | sum | agg | ebuf  (~136 MB)
    float* ws = (float*)d_ws;
    size_t off = 0;
    float*    h       = ws + off; off += (size_t)N_NODES * HC;
    float*    a_src_n = ws + off; off += (size_t)N_NODES * HEADS;
    float*    a_dst_n = ws + off; off += (size_t)N_NODES * HEADS;
    unsigned* mbuf    = (unsigned*)(ws + off); off += (size_t)N_NODES * HEADS;
    float*    sumbuf  = ws + off; off += (size_t)N_NODES * HEADS;
    float*    agg     = ws + off; off += (size_t)N_NODES * HC;
    float*    ebuf    = ws + off; off += (size_t)Etot * HEADS;

    init_kernel<<<(N_NODES * HC + 255) / 256, 256, 0, stream>>>(agg, sumbuf, mbuf);

    gemm_x_proj<<<6250, 256, 0, stream>>>(x, W, h);

    node_attn<<<(N_NODES * HEADS + 255) / 256, 256, 0, stream>>>(h, att_src, att_dst,
                                                                 a_src_n, a_dst_n);

    int eg = (Etot + 255) / 256;
    edge_logits_max<<<eg, 256, 0, stream>>>(ei, E, Etot, a_src_n, a_dst_n, ebuf, mbuf);
    edge_exp_sum  <<<eg, 256, 0, stream>>>(ei, E, Etot, mbuf, ebuf, sumbuf);
    edge_alpha    <<<eg, 256, 0, stream>>>(ei, E, Etot, sumbuf, ebuf);

    long long aggThreads = (long long)Etot * (HC / 4);
    edge_aggregate<<<(unsigned)((aggThreads + 255) / 256), 256, 0, stream>>>(ei, E, Etot,
                                                                             ebuf, h, agg);

    gemm_out_elu<<<(12500 + 7) / 8, 256, 0, stream>>>(agg, bias_conv, W_lin, b_lin, out);
}